// PointGroup_3753801416972
// MI455X (gfx1250) — compile-verified
//
#include <hip/hip_runtime.h>

// ---------------------------------------------------------------------------
// Point grouping for MI455X (gfx1250, wave32):
//   FPS (512 sequential argmax rounds per batch)  -> centers
//   KNN top-32 via V_WMMA_F32_16X16X4_F32 scores  -> neighborhoods
//
// sqd(c,p) = c^2 + p^2 - 2 c.p is computed as one rank-4 WMMA:
//   A(16x4)  = [-2cx, -2cy, -2cz, 1]   (per center row)
//   B(4x16)  = [ x ;  y ;  z ; p^2 ]   (per point column)
//   C seed   = c^2 (broadcast per row) -> D = full sqd tile, fp32.
//
// FPS cross-wave argmax: dist >= 0 so f32 bits are order-monotonic; pack
// key = (bits(dist)<<32) | ~idx and ds_max_u64 from each wave leader.
// max(key) == lexicographic (larger dist, tie -> smaller idx) == jnp.argmax.
// ---------------------------------------------------------------------------

typedef __attribute__((ext_vector_type(2))) float v2f;
typedef __attribute__((ext_vector_type(8))) float v8f;

#define BATCH     32
#define NPTS      8192
#define NGRP      512
#define GSZ       32
#define FEAT      6
#define CHUNK     512          // points per KNN chunk staged in LDS
#define SROW      514          // padded score-row stride (bank-conflict free)

__device__ __forceinline__ int bperm_i(int src, int l) {
    return __builtin_amdgcn_ds_bpermute(l << 2, src);
}
__device__ __forceinline__ float bperm_f(float src, int l) {
    return __int_as_float(bperm_i(__float_as_int(src), l));
}

// (v1,i1) strictly precedes (v2,i2): smaller value, tie -> smaller index.
__device__ __forceinline__ bool lexLess(float v1, int i1, float v2, int i2) {
    return (v1 < v2) || (v1 == v2 && i1 < i2);
}
// argmax order: larger value wins, tie -> smaller index (matches jnp.argmax).
__device__ __forceinline__ bool lexGreater(float v1, int i1, float v2, int i2) {
    return (v1 > v2) || (v1 == v2 && i1 < i2);
}

__device__ __forceinline__ unsigned long long packKey(float d, int idx) {
    return ((unsigned long long)(unsigned)__float_as_int(d) << 32) |
           (unsigned)(~idx);
}

// ---------------------------------------------------------------------------
// Kernel 1: furthest point sampling. One block (1024 threads = 32 waves)
// per batch; 8 points per thread live in registers for all 512 rounds.
// Per round: compute + wave butterfly + 32 leader ds_max_u64, 2 barriers.
// ---------------------------------------------------------------------------
__global__ __launch_bounds__(1024) void fps_kernel(const float* __restrict__ pc,
                                                   float* centers) {
    const int b    = blockIdx.x;
    const int t    = threadIdx.x;
    const int lane = t & 31;

    __shared__ unsigned long long keys[2];   // double-buffered argmax key
    __shared__ float cur[3];

    const float* pcb = pc + (size_t)b * NPTS * FEAT;

    float px[8], py[8], pz[8], mind[8];
#pragma unroll
    for (int k = 0; k < 8; ++k) {
        const int p = t + (k << 10);
        const float* q = pcb + (size_t)p * FEAT;
        px[k] = q[0]; py[k] = q[1]; pz[k] = q[2];
        mind[k] = __builtin_inff();
    }
    if (t == 0) {
        cur[0] = pcb[0]; cur[1] = pcb[1]; cur[2] = pcb[2];
        keys[0] = 0ull;  keys[1] = 0ull;
    }
    __syncthreads();

    for (int i = 0; i < NGRP; ++i) {
        const float cx = cur[0], cy = cur[1], cz = cur[2];
        if (t == 0) {
            float* o = centers + ((size_t)b * NGRP + i) * 3;
            o[0] = cx; o[1] = cy; o[2] = cz;
            keys[(i + 1) & 1] = 0ull;        // reset other buffer (race-free)
        }
        // update min-dist, local argmax (ascending k == ascending global idx)
        float best = -1.0f;
        int   bidx = 0x7fffffff;
#pragma unroll
        for (int k = 0; k < 8; ++k) {
            const float dx = px[k] - cx, dy = py[k] - cy, dz = pz[k] - cz;
            const float d  = dx * dx + dy * dy + dz * dz;
            const float md = fminf(mind[k], d);
            mind[k] = md;
            if (md > best) { best = md; bidx = t + (k << 10); }
        }
        // wave32 butterfly argmax
#pragma unroll
        for (int mm = 16; mm >= 1; mm >>= 1) {
            const float ov = bperm_f(best, lane ^ mm);
            const int   oi = bperm_i(bidx, lane ^ mm);
            if (lexGreater(ov, oi, best, bidx)) { best = ov; bidx = oi; }
        }
        if (lane == 0)
            atomicMax(&keys[i & 1], packKey(best, bidx));
        __syncthreads();                     // atomics visible to everyone
        const int si = ~(unsigned)(keys[i & 1] & 0xffffffffull);
        if (t == (si & 1023)) {              // owner publishes winner coords
            const int k = si >> 10;
            cur[0] = px[k]; cur[1] = py[k]; cur[2] = pz[k];
        }
        __syncthreads();                     // cur ready for next round
    }
}

// ---------------------------------------------------------------------------
// Kernel 2: KNN (top-32 smallest sqd) + gather. One block (512 threads =
// 16 waves) per (batch, 16-center tile). Wave w owns center g0+w's sorted
// top-32 list in registers (one (val,idx) pair per lane).
// ---------------------------------------------------------------------------
__global__ __launch_bounds__(512) void knn_kernel(const float* __restrict__ pc,
                                                  float* out) {
    __shared__ float  sc[16 * SROW];   // score tile, padded rows
    __shared__ float4 pts[CHUNK];      // (x, y, z, |p|^2)
    __shared__ float  ctr[16 * 3];

    const int b    = blockIdx.x >> 5;
    const int tile = blockIdx.x & 31;
    const int g0   = tile * 16;
    const int t    = threadIdx.x;
    const int lane = t & 31;
    const int wv   = t >> 5;
    const int m    = lane & 15;

    const float* pcb   = pc  + (size_t)b * NPTS * FEAT;
    float*       nbOut = out;
    const float* cenIn = out + (size_t)BATCH * NGRP * GSZ * FEAT;

    if (t < 48) ctr[t] = cenIn[((size_t)b * NGRP + g0) * 3 + t];
    __syncthreads();

    // A operand (16x4 f32): lanes 0-15 -> K0,K1 ; lanes 16-31 -> K2,K3
    v2f A;
    {
        const float cx = ctr[m * 3 + 0], cy = ctr[m * 3 + 1], cz = ctr[m * 3 + 2];
        A.x = (lane < 16) ? (-2.0f * cx) : (-2.0f * cz);
        A.y = (lane < 16) ? (-2.0f * cy) : 1.0f;
    }
    // C seed: VGPR v holds row v (lanes 0-15) / row v+8 (lanes 16-31) -> c^2
    v8f Cs;
#pragma unroll
    for (int v = 0; v < 8; ++v) {
        const int mm = (lane < 16) ? v : (v + 8);
        const float ax = ctr[mm * 3], ay = ctr[mm * 3 + 1], az = ctr[mm * 3 + 2];
        Cs[v] = ax * ax + ay * ay + az * az;
    }

    // sorted top-32 list: lane j holds the j-th smallest (val, idx)
    float lv = __builtin_inff();
    int   li = 0x7fffffff;

    for (int c = 0; c < NPTS / CHUNK; ++c) {
        const int base = c * CHUNK;
        __syncthreads();               // prior selection done reading sc/pts
        {                              // stage one point per thread
            const float* q = pcb + (size_t)(base + t) * FEAT;
            const float x = q[0], y = q[1], z = q[2];
            pts[t] = make_float4(x, y, z, x * x + y * y + z * z);
        }
        __syncthreads();

        // wave w computes N-tiles {w, w+16} with f32 WMMA
#pragma unroll
        for (int tt = 0; tt < 2; ++tt) {
            const int nt = wv + tt * 16;
            const float4 P = pts[nt * 16 + m];
            v2f B;
            B.x = (lane < 16) ? P.x : P.z;
            B.y = (lane < 16) ? P.y : P.w;
            v8f D = __builtin_amdgcn_wmma_f32_16x16x4_f32(
                false, A, false, B, (short)0, Cs, false, false);
#pragma unroll
            for (int v = 0; v < 8; ++v) {
                const int mm = (lane < 16) ? v : (v + 8);
                sc[mm * SROW + nt * 16 + m] = D[v];
            }
        }
        __syncthreads();

        // wave w merges its center's CHUNK scores into the sorted top-32
#pragma unroll 1
        for (int cc = 0; cc < CHUNK / 32; ++cc) {
            const float val = sc[wv * SROW + cc * 32 + lane];
            const int   idx = base + cc * 32 + lane;
            float tv = bperm_f(lv, 31);
            int   ti = bperm_i(li, 31);
            unsigned msk = __builtin_amdgcn_ballot_w32(lexLess(val, idx, tv, ti));
            while (msk) {
                const int s = __builtin_ctz(msk);
                msk &= msk - 1;
                const float cvv = bperm_f(val, s);
                const int   cii = bperm_i(idx, s);
                tv = bperm_f(lv, 31);
                ti = bperm_i(li, 31);
                if (lexLess(cvv, cii, tv, ti)) {
                    const unsigned pm =
                        __builtin_amdgcn_ballot_w32(lexLess(lv, li, cvv, cii));
                    const int p = __builtin_popcount(pm);
                    const float sv = bperm_f(lv, lane - 1);
                    const int   si = bperm_i(li, lane - 1);
                    if (lane >= p) {
                        lv = (lane == p) ? cvv : sv;
                        li = (lane == p) ? cii : si;
                    }
                }
            }
        }
    }

    // gather: wave w writes neighborhoods[b][g0+w][lane][0..5]
    {
        const float ccx = ctr[wv * 3], ccy = ctr[wv * 3 + 1], ccz = ctr[wv * 3 + 2];
        const float* q = pcb + (size_t)li * FEAT;
        float* o = nbOut + (((size_t)(b * NGRP + g0 + wv) * GSZ) + lane) * FEAT;
        o[0] = q[0] - ccx;
        o[1] = q[1] - ccy;
        o[2] = q[2] - ccz;
        o[3] = q[3];
        o[4] = q[4];
        o[5] = q[5];
    }
}

extern "C" void kernel_launch(void* const* d_in, const int* in_sizes, int n_in,
                              void* d_out, int out_size, void* d_ws, size_t ws_size,
                              hipStream_t stream) {
    (void)in_sizes; (void)n_in; (void)out_size; (void)d_ws; (void)ws_size;
    const float* pc = (const float*)d_in[0];
    float* out = (float*)d_out;
    float* centers = out + (size_t)BATCH * NGRP * GSZ * FEAT;  // tail of d_out

    fps_kernel<<<BATCH, 1024, 0, stream>>>(pc, centers);
    knn_kernel<<<BATCH * (NGRP / 16), 512, 0, stream>>>(pc, out);
}